// BernConvLayer_592705487393
// MI455X (gfx1250) — compile-verified
//
#include <hip/hip_runtime.h>
#include <hip/hip_bf16.h>

// ---------------------------------------------------------------------------
// BernNet (Bernstein spectral GCN) for MI455X / gfx1250, fp32 via
// v_wmma_f32_16x16x4_f32.  LDS-tiled, double-buffered WMMA GEMM; the
// normalized adjacency Ahat = D^-1/2 A D^-1/2 is fused into the A-tile load.
// B is stored transposed in LDS so both A and B fragments are single
// contiguous ds_load_b64 ops.
// ---------------------------------------------------------------------------

typedef __attribute__((ext_vector_type(2))) float v2f;
typedef __attribute__((ext_vector_type(8))) float v8f;

#define TM 128        // workgroup tile M
#define TN 128        // workgroup tile N
#define TK 32         // K tile
#define LDS_S 34      // row stride (floats): 32 + pad, even -> aligned b64, bank-spread

__device__ __forceinline__ v8f wmma4(v2f a, v2f b, v8f c) {
  // D = A(16x4 f32) * B(4x16 f32) + C(16x16 f32)
  return __builtin_amdgcn_wmma_f32_16x16x4_f32(
      /*neg_a=*/false, a, /*neg_b=*/false, b,
      /*c_mod=*/(short)0, c, /*reuse_a=*/false, /*reuse_b=*/false);
}

// Fragment lane layouts per CDNA5 ISA 7.12.2:
//  A 16x4 : lanes 0-15 -> {K0,K1}, lanes 16-31 -> {K2,K3}, M = lane&15
//  B 4x16 : lanes 0-15 -> {K0,K1}, lanes 16-31 -> {K2,K3}, N = lane&15
//  C 16x16: vgpr g -> M = g + 8*(lane>=16), N = lane&15
// As  : [row][k]  row-major, stride LDS_S
// BsT : [col][k]  (transposed!) stride LDS_S -> B frag is contiguous pair
__device__ __forceinline__ void mma_ktile(const float* As, const float* BsT,
                                          v8f acc[4][2], int lane, int wmW, int wnW) {
  const int lo = lane & 15;
  const int hi2 = (lane >> 4) << 1;  // 0 or 2
#pragma unroll
  for (int kk = 0; kk < TK / 4; ++kk) {
    const int kb = kk * 4 + hi2;
    v2f afr[4], bfr[2];
#pragma unroll
    for (int mi = 0; mi < 4; ++mi) {
      const float* p = &As[(wmW * 64 + mi * 16 + lo) * LDS_S + kb];
      v2f a; a.x = p[0]; a.y = p[1];
      afr[mi] = a;
    }
#pragma unroll
    for (int ni = 0; ni < 2; ++ni) {
      const float* p = &BsT[(wnW * 32 + ni * 16 + lo) * LDS_S + kb];
      v2f b; b.x = p[0]; b.y = p[1];
      bfr[ni] = b;
    }
#pragma unroll
    for (int mi = 0; mi < 4; ++mi)
#pragma unroll
      for (int ni = 0; ni < 2; ++ni)
        acc[mi][ni] = wmma4(afr[mi], bfr[ni], acc[mi][ni]);
  }
}

// ---------------------------------------------------------------------------
// spectral_gemm: out = s_mm * (Ahat @ Vin) + a1*X1 + cvec[cidx]*X2  (+ReLU)
// Ahat built on the fly from adj (>0 mask) and dis (per-row rsqrt degree).
// Shapes per batch (blockIdx.z): Ahat [n,n], Vin/X1/X2/out [n,hd].
// ---------------------------------------------------------------------------
__global__ __launch_bounds__(256) void spectral_gemm(
    float* __restrict__ out, const float* __restrict__ adj,
    const float* __restrict__ dis, const float* __restrict__ V,
    const float* __restrict__ X1, float a1,
    const float* __restrict__ X2, const float* __restrict__ cvec, int cidx,
    float s_mm, int relu, int n, int hd) {
  __shared__ float As[2][TM * LDS_S];
  __shared__ float Bt[2][TN * LDS_S];

  const int b = blockIdx.z;
  const float* adjb = adj + (size_t)b * n * n;
  const float* disb = dis + (size_t)b * n;
  const size_t vOff = (size_t)b * n * hd;

  const int m0 = blockIdx.x * TM;
  const int n0 = blockIdx.y * TN;
  const int t = threadIdx.x;
  const int lane = t & 31, wave = t >> 5;
  const int wmW = wave >> 2, wnW = wave & 3;  // 2 x 4 waves -> 64x32 per wave

  const v8f vzero = {0.f, 0.f, 0.f, 0.f, 0.f, 0.f, 0.f, 0.f};
  v8f acc[4][2] = {{vzero, vzero}, {vzero, vzero}, {vzero, vzero}, {vzero, vzero}};

  float4 aR[4], bR[4];

  auto loadTile = [&](int k0) {
    // A: 128 x 32 of Ahat (fused normalization)
#pragma unroll
    for (int it = 0; it < 4; ++it) {
      const int slot = t + it * 256;         // 1024 float4 slots
      const int row = slot >> 3;             // 0..127
      const int cg = (slot & 7) << 2;        // 0..28
      const float4 av = *(const float4*)&adjb[(size_t)(m0 + row) * n + k0 + cg];
      const float4 dv = *(const float4*)&disb[k0 + cg];
      const float dr = disb[m0 + row];
      aR[it].x = (av.x > 0.f) ? dr * dv.x : 0.f;
      aR[it].y = (av.y > 0.f) ? dr * dv.y : 0.f;
      aR[it].z = (av.z > 0.f) ? dr * dv.z : 0.f;
      aR[it].w = (av.w > 0.f) ? dr * dv.w : 0.f;
    }
    // B: 32 x 128 of Vin
#pragma unroll
    for (int it = 0; it < 4; ++it) {
      const int slot = t + it * 256;         // 1024 float4 slots
      const int row = slot >> 5;             // 0..31 (k)
      const int cg = (slot & 31) << 2;       // 0..124 (col)
      bR[it] = *(const float4*)&V[vOff + (size_t)(k0 + row) * hd + n0 + cg];
    }
  };

  auto storeTile = [&](int bi) {
#pragma unroll
    for (int it = 0; it < 4; ++it) {
      const int slot = t + it * 256;
      const int row = slot >> 3;
      const int cg = (slot & 7) << 2;
      float* dp = &As[bi][row * LDS_S + cg];
      dp[0] = aR[it].x; dp[1] = aR[it].y; dp[2] = aR[it].z; dp[3] = aR[it].w;
    }
#pragma unroll
    for (int it = 0; it < 4; ++it) {
      const int slot = t + it * 256;
      const int row = slot >> 5;             // k
      const int cg = (slot & 31) << 2;       // col
      Bt[bi][(cg + 0) * LDS_S + row] = bR[it].x;
      Bt[bi][(cg + 1) * LDS_S + row] = bR[it].y;
      Bt[bi][(cg + 2) * LDS_S + row] = bR[it].z;
      Bt[bi][(cg + 3) * LDS_S + row] = bR[it].w;
    }
  };

  loadTile(0);
  storeTile(0);
  __syncthreads();

  const int nK = n / TK;
  for (int kt = 0; kt < nK; ++kt) {
    const int cur = kt & 1;
    if (kt + 1 < nK) loadTile((kt + 1) * TK);
    mma_ktile(As[cur], Bt[cur], acc, lane, wmW, wnW);
    if (kt + 1 < nK) storeTile(cur ^ 1);
    __syncthreads();
  }

  const float a2 = (cidx >= 0) ? cvec[cidx] : 0.f;
  const int lo = lane & 15, hi = lane >> 4;
#pragma unroll
  for (int mi = 0; mi < 4; ++mi)
#pragma unroll
    for (int ni = 0; ni < 2; ++ni)
#pragma unroll
      for (int g = 0; g < 8; ++g) {
        const int row = m0 + wmW * 64 + mi * 16 + g + hi * 8;
        const int col = n0 + wnW * 32 + ni * 16 + lo;
        const size_t o = vOff + (size_t)row * hd + col;
        float v = s_mm * acc[mi][ni][g];
        if (X1) v += a1 * X1[o];
        if (cidx >= 0) v += a2 * X2[o];
        if (relu) v = fmaxf(v, 0.f);
        out[o] = v;
      }
}

// ---------------------------------------------------------------------------
// linear_gemm: out[M,N] = X[M,K] @ W[K,N] + bias[N]   (optional ReLU)
// ---------------------------------------------------------------------------
__global__ __launch_bounds__(256) void linear_gemm(
    float* __restrict__ out, const float* __restrict__ X,
    const float* __restrict__ W, const float* __restrict__ bias,
    int M, int Kdim, int N, int relu) {
  __shared__ float As[2][TM * LDS_S];
  __shared__ float Bt[2][TN * LDS_S];

  const int m0 = blockIdx.x * TM;
  const int n0 = blockIdx.y * TN;
  const int t = threadIdx.x;
  const int lane = t & 31, wave = t >> 5;
  const int wmW = wave >> 2, wnW = wave & 3;

  const v8f vzero = {0.f, 0.f, 0.f, 0.f, 0.f, 0.f, 0.f, 0.f};
  v8f acc[4][2] = {{vzero, vzero}, {vzero, vzero}, {vzero, vzero}, {vzero, vzero}};

  float4 aR[4], bR[4];

  auto loadTile = [&](int k0) {
#pragma unroll
    for (int it = 0; it < 4; ++it) {
      const int slot = t + it * 256;
      const int row = slot >> 3;
      const int cg = (slot & 7) << 2;
      aR[it] = *(const float4*)&X[(size_t)(m0 + row) * Kdim + k0 + cg];
    }
#pragma unroll
    for (int it = 0; it < 4; ++it) {
      const int slot = t + it * 256;
      const int row = slot >> 5;
      const int cg = (slot & 31) << 2;
      bR[it] = *(const float4*)&W[(size_t)(k0 + row) * N + n0 + cg];
    }
  };

  auto storeTile = [&](int bi) {
#pragma unroll
    for (int it = 0; it < 4; ++it) {
      const int slot = t + it * 256;
      const int row = slot >> 3;
      const int cg = (slot & 7) << 2;
      float* dp = &As[bi][row * LDS_S + cg];
      dp[0] = aR[it].x; dp[1] = aR[it].y; dp[2] = aR[it].z; dp[3] = aR[it].w;
    }
#pragma unroll
    for (int it = 0; it < 4; ++it) {
      const int slot = t + it * 256;
      const int row = slot >> 5;
      const int cg = (slot & 31) << 2;
      Bt[bi][(cg + 0) * LDS_S + row] = bR[it].x;
      Bt[bi][(cg + 1) * LDS_S + row] = bR[it].y;
      Bt[bi][(cg + 2) * LDS_S + row] = bR[it].z;
      Bt[bi][(cg + 3) * LDS_S + row] = bR[it].w;
    }
  };

  loadTile(0);
  storeTile(0);
  __syncthreads();

  const int nK = Kdim / TK;
  for (int kt = 0; kt < nK; ++kt) {
    const int cur = kt & 1;
    if (kt + 1 < nK) loadTile((kt + 1) * TK);
    mma_ktile(As[cur], Bt[cur], acc, lane, wmW, wnW);
    if (kt + 1 < nK) storeTile(cur ^ 1);
    __syncthreads();
  }

  const int lo = lane & 15, hi = lane >> 4;
#pragma unroll
  for (int mi = 0; mi < 4; ++mi)
#pragma unroll
    for (int ni = 0; ni < 2; ++ni)
#pragma unroll
      for (int g = 0; g < 8; ++g) {
        const int row = m0 + wmW * 64 + mi * 16 + g + hi * 8;
        const int col = n0 + wnW * 32 + ni * 16 + lo;
        float v = acc[mi][ni][g];
        if (bias) v += bias[col];
        if (relu) v = fmaxf(v, 0.f);
        out[(size_t)row * N + col] = v;
      }
}

// ---------------------------------------------------------------------------
// Small helper kernels
// ---------------------------------------------------------------------------
__global__ void norm_deg_kernel(const float* __restrict__ adj,
                                float* __restrict__ dis, int n) {
  const int row = blockIdx.x;  // over B*n rows
  const size_t base = (size_t)row * n;
  float s = 0.f;
  for (int c = threadIdx.x; c < n; c += 256)
    s += (adj[base + c] > 0.f) ? 1.f : 0.f;
  __shared__ float red[256];
  red[threadIdx.x] = s;
  __syncthreads();
  for (int off = 128; off > 0; off >>= 1) {
    if (threadIdx.x < off) red[threadIdx.x] += red[threadIdx.x + off];
    __syncthreads();
  }
  if (threadIdx.x == 0) {
    const float d = red[0];
    dis[row] = (d > 0.f) ? rsqrtf(d) : 0.f;
  }
}

__global__ void cvec_kernel(const float* __restrict__ coe, float* __restrict__ cvec) {
  const int t = threadIdx.x;
  const float binom[11] = {1.f, 10.f, 45.f, 120.f, 210.f, 252.f,
                           210.f, 120.f, 45.f, 10.f, 1.f};
  if (t < 11) cvec[t] = (binom[t] * (1.f / 1024.f)) * fmaxf(coe[t], 0.f);
}

__global__ void scale_kernel(float* __restrict__ out, const float* __restrict__ src,
                             const float* __restrict__ cvec, int cidx, long long total) {
  const long long i = (long long)blockIdx.x * blockDim.x + threadIdx.x;
  if (i < total) out[i] = cvec[cidx] * src[i];
}

__global__ void copy_cols_kernel(float* __restrict__ dst, int dstW, int dstOff,
                                 const float* __restrict__ src, int srcW,
                                 long long total) {
  const long long i = (long long)blockIdx.x * blockDim.x + threadIdx.x;
  if (i >= total) return;
  const long long r = i / srcW;
  const int c = (int)(i - r * srcW);
  dst[r * dstW + dstOff + c] = src[i];
}

__global__ void bern_kernel(float* __restrict__ dst, const float* __restrict__ x,
                            const float* __restrict__ h0, const float* __restrict__ h1,
                            const float* __restrict__ h2, long long total) {
  const long long i = (long long)blockIdx.x * blockDim.x + threadIdx.x;
  if (i >= total) return;
  const long long r = i / 768;
  const int c = (int)(i - r * 768);
  const int sel = c >> 8;  // hd = 256
  const int cc = c & 255;
  const float* h = (sel == 0) ? h0 : ((sel == 1) ? h1 : h2);
  dst[i] = h[r * 256 + cc] + x[i];
}

// ---------------------------------------------------------------------------
// Orchestration
// ---------------------------------------------------------------------------
extern "C" void kernel_launch(void* const* d_in, const int* in_sizes, int n_in,
                              void* d_out, int out_size, void* d_ws, size_t ws_size,
                              hipStream_t stream) {
  const float* adj  = (const float*)d_in[0];
  const float* x    = (const float*)d_in[1];
  const float* coe  = (const float*)d_in[2];
  const float* W0   = (const float*)d_in[3];
  const float* b0   = (const float*)d_in[4];
  const float* W1   = (const float*)d_in[5];
  const float* b1   = (const float*)d_in[6];
  const float* W2   = (const float*)d_in[7];
  const float* b2   = (const float*)d_in[8];
  const float* Wout = (const float*)d_in[9];
  const float* bout = (const float*)d_in[10];
  float* out = (float*)d_out;

  const int B = 8, n = 2048, H = 768, hd = 256, Kord = 10;
  const long long Mrows = (long long)B * n;     // 16384
  const size_t S = (size_t)B * n * hd;          // per-feature-map elements

  // workspace layout (floats)
  float* w = (float*)d_ws;
  float* dis  = w;  w += (size_t)B * n;
  float* cv   = w;  w += 16;
  float* tmp  = w;  w += (size_t)(Kord + 1) * S;   // tmp[0..K]
  float* accA = w;  w += S;
  float* accB = w;  w += S;
  float* h0   = w;  w += S;
  float* h1   = w;  w += S;
  float* h2   = w;  w += S;
  float* xcat = w;                                  // up to B*n*1280, reused for bern

  const dim3 blk(256);
  const dim3 sgrid(n / TM, hd / TN, B);

  norm_deg_kernel<<<dim3(B * n), blk, 0, stream>>>(adj, dis, n);
  cvec_kernel<<<1, 32, 0, stream>>>(coe, cv);

  auto run_layer = [&](const float* Xin, int Kin, const float* Wl,
                       const float* bl, float* hout) {
    // h = Xin @ Wl + bl  -> tmp[0]
    linear_gemm<<<dim3((unsigned)(Mrows / TM), hd / TN, 1), blk, 0, stream>>>(
        tmp, Xin, Wl, bl, (int)Mrows, Kin, hd, 0);
    // powers of (I + Ahat)
    for (int j = 1; j <= Kord; ++j)
      spectral_gemm<<<sgrid, blk, 0, stream>>>(
          tmp + (size_t)j * S, adj, dis, tmp + (size_t)(j - 1) * S,
          tmp + (size_t)(j - 1) * S, 1.f, nullptr, cv, -1, 1.f, 0, n, hd);
    // acc = c[K] * tmp[0]
    scale_kernel<<<dim3((unsigned)((S + 255) / 256)), blk, 0, stream>>>(
        accA, tmp, cv, Kord, (long long)S);
    // Horner: acc = (acc - Ahat@acc) + c[i]*tmp[K-i]
    float* ai = accA; float* ao = accB;
    for (int i = Kord - 1; i >= 1; --i) {
      spectral_gemm<<<sgrid, blk, 0, stream>>>(
          ao, adj, dis, ai, ai, 1.f, tmp + (size_t)(Kord - i) * S, cv, i,
          -1.f, 0, n, hd);
      float* t_ = ai; ai = ao; ao = t_;
    }
    // h_layer = relu(c[0]*tmp[K] + acc - Ahat@acc)
    spectral_gemm<<<sgrid, blk, 0, stream>>>(
        hout, adj, dis, ai, ai, 1.f, tmp + (size_t)Kord * S, cv, 0,
        -1.f, 1, n, hd);
  };

  // ---- Layer 0: input x (K = 768) ----
  run_layer(x, H, W0, b0, h0);

  // ---- Layer 1: [x | h0] (K = 1024) ----
  {
    const long long tx = Mrows * (long long)H;
    copy_cols_kernel<<<dim3((unsigned)((tx + 255) / 256)), blk, 0, stream>>>(
        xcat, H + hd, 0, x, H, tx);
    const long long th = Mrows * (long long)hd;
    copy_cols_kernel<<<dim3((unsigned)((th + 255) / 256)), blk, 0, stream>>>(
        xcat, H + hd, H, h0, hd, th);
  }
  run_layer(xcat, H + hd, W1, b1, h1);

  // ---- Layer 2: [x | h0 | h1] (K = 1280) ----
  {
    const long long tx = Mrows * (long long)H;
    copy_cols_kernel<<<dim3((unsigned)((tx + 255) / 256)), blk, 0, stream>>>(
        xcat, H + 2 * hd, 0, x, H, tx);
    const long long th = Mrows * (long long)hd;
    copy_cols_kernel<<<dim3((unsigned)((th + 255) / 256)), blk, 0, stream>>>(
        xcat, H + 2 * hd, H, h0, hd, th);
    copy_cols_kernel<<<dim3((unsigned)((th + 255) / 256)), blk, 0, stream>>>(
        xcat, H + 2 * hd, H + hd, h1, hd, th);
  }
  run_layer(xcat, H + 2 * hd, W2, b2, h2);

  // ---- bern = concat(h0,h1,h2) + x ; out = bern @ Wout + bout ----
  {
    const long long tb = Mrows * (long long)H;
    bern_kernel<<<dim3((unsigned)((tb + 255) / 256)), blk, 0, stream>>>(
        xcat, x, h0, h1, h2, tb);
  }
  linear_gemm<<<dim3((unsigned)(Mrows / TM), H / TN, 1), blk, 0, stream>>>(
      out, xcat, Wout, bout, (int)Mrows, H, H, 0);
}